// MessageGcn_206158430367
// MI455X (gfx1250) — compile-verified
//
#include <hip/hip_runtime.h>
#include <hip/hip_bf16.h>

typedef __attribute__((ext_vector_type(2))) float v2f;
typedef __attribute__((ext_vector_type(8))) float v8f;

#define D 128
#define LDSPAD 132   // 132 floats = 528 B row stride: multiple of 16 B, bank-conflict-free frags

// Fused triple-GEMM: for one 16-row tile of x, compute
//   Mf = x @ W_f,  Mb = x @ W_b,  out = (x @ W_s) * dropout_mask
// reading the x tile from HBM exactly once (staged in LDS).
// 128 threads = 4 waves; wave w owns columns [32w, 32w+32) as two 16x16 tiles,
// for all three weight matrices => 6 WMMA accumulators per wave.
__global__ __launch_bounds__(128) void gemm16_wmma_x3(
    const float* __restrict__ A,
    const float* __restrict__ Wf, const float* __restrict__ Wb, const float* __restrict__ Ws,
    const float* __restrict__ dropu,
    float* __restrict__ Mf, float* __restrict__ Mb, float* __restrict__ out)
{
    __shared__ float sA[16 * LDSPAD];
    const int t = threadIdx.x;
    const size_t rowBase = (size_t)blockIdx.x * 16;

    // Stage the 16x128 A tile: 4 rows in flight, 32 lanes x float4 per row (coalesced).
    {
        const int r0 = t >> 5;          // 0..3
        const int c4 = (t & 31) * 4;    // 0,4,...,124
#pragma unroll
        for (int it = 0; it < 4; ++it) {
            const int row = it * 4 + r0;
            const float4 v = *(const float4*)(A + (rowBase + row) * D + c4);
            *(float4*)(&sA[row * LDSPAD + c4]) = v;
        }
    }
    __syncthreads();

    const int wave = t >> 5;        // 0..3
    const int lane = t & 31;
    const int half = lane >> 4;     // 0: lanes 0-15, 1: lanes 16-31
    const int m16  = lane & 15;
    const int colA = wave * 32 + m16;        // B-frag / C column, tile 0
    const int colB = colA + 16;              // tile 1

    v8f cf0 = {}, cf1 = {};
    v8f cb0 = {}, cb1 = {};
    v8f cs0 = {}, cs1 = {};

#pragma unroll 2
    for (int k = 0; k < D; k += 4) {
        // A frag (ISA 32-bit 16x4 layout): lane half selects K pair {k+2h, k+2h+1}.
        const v2f a = *(const v2f*)(&sA[m16 * LDSPAD + k + 2 * half]);
        const int kr = k + 2 * half;
        const size_t o0 = (size_t)kr * D;
        const size_t o1 = (size_t)(kr + 1) * D;

        v2f bf0, bf1, bb0, bb1, bs0, bs1;
        bf0.x = Wf[o0 + colA];  bf0.y = Wf[o1 + colA];
        bf1.x = Wf[o0 + colB];  bf1.y = Wf[o1 + colB];
        bb0.x = Wb[o0 + colA];  bb0.y = Wb[o1 + colA];
        bb1.x = Wb[o0 + colB];  bb1.y = Wb[o1 + colB];
        bs0.x = Ws[o0 + colA];  bs0.y = Ws[o1 + colA];
        bs1.x = Ws[o0 + colB];  bs1.y = Ws[o1 + colB];

        // (neg_a, A, neg_b, B, c_mod, C, reuse_a, reuse_b)
        cf0 = __builtin_amdgcn_wmma_f32_16x16x4_f32(false, a, false, bf0, (short)0, cf0, false, false);
        cf1 = __builtin_amdgcn_wmma_f32_16x16x4_f32(false, a, false, bf1, (short)0, cf1, false, false);
        cb0 = __builtin_amdgcn_wmma_f32_16x16x4_f32(false, a, false, bb0, (short)0, cb0, false, false);
        cb1 = __builtin_amdgcn_wmma_f32_16x16x4_f32(false, a, false, bb1, (short)0, cb1, false, false);
        cs0 = __builtin_amdgcn_wmma_f32_16x16x4_f32(false, a, false, bs0, (short)0, cs0, false, false);
        cs1 = __builtin_amdgcn_wmma_f32_16x16x4_f32(false, a, false, bs1, (short)0, cs1, false, false);
    }

    // C layout: VGPR j -> M = j + 8*half, N = lane&15 within the 16-col tile.
    const int mOff = half * 8;
#pragma unroll
    for (int j = 0; j < 8; ++j) {
        const size_t row = rowBase + mOff + j;
        Mf[row * D + colA] = cf0[j];
        Mf[row * D + colB] = cf1[j];
        Mb[row * D + colA] = cb0[j];
        Mb[row * D + colB] = cb1[j];
        // self-loop with fused inverted dropout (1/0.8 == 1.25 exactly)
        const float mA = (dropu[row * D + colA] < 0.8f) ? 1.25f : 0.0f;
        const float mB = (dropu[row * D + colB] < 0.8f) ? 1.25f : 0.0f;
        out[row * D + colA] = cs0[j] * mA;
        out[row * D + colB] = cs1[j] * mB;
    }
}

// Pure data-movement edge phase: out[r] += Mf[s]; out[s] += Mb[r].
// One wave per edge (lanes share the index loads); each lane owns a float4 column group,
// so reads are full coalesced 512B rows and atomics are contiguous global_atomic_add_f32
// (no-return form, STOREcnt-tracked) into the L2-resident output.
__global__ __launch_bounds__(256) void edge_scatter(
    const float* __restrict__ Mf, const float* __restrict__ Mb,
    const int* __restrict__ senders, const int* __restrict__ receivers,
    float* __restrict__ out, int E)
{
    const long long idx = (long long)blockIdx.x * blockDim.x + threadIdx.x;
    const int e = (int)(idx >> 5);
    if (e >= E) return;
    const int q = (int)(idx & 31) * 4;      // column group 0,4,...,124

    const int s = senders[e];
    const int r = receivers[e];

    const float4 mf = *(const float4*)(Mf + (size_t)s * D + q);
    const float4 mb = *(const float4*)(Mb + (size_t)r * D + q);

    float* outR = out + (size_t)r * D + q;
    float* outS = out + (size_t)s * D + q;
    atomicAdd(outR + 0, mf.x);
    atomicAdd(outR + 1, mf.y);
    atomicAdd(outR + 2, mf.z);
    atomicAdd(outR + 3, mf.w);
    atomicAdd(outS + 0, mb.x);
    atomicAdd(outS + 1, mb.y);
    atomicAdd(outS + 2, mb.z);
    atomicAdd(outS + 3, mb.w);
}

__global__ __launch_bounds__(256) void relu_inplace(float* __restrict__ out, long long n4)
{
    const long long i = (long long)blockIdx.x * blockDim.x + threadIdx.x;
    if (i >= n4) return;
    float4 v = ((float4*)out)[i];
    v.x = fmaxf(v.x, 0.0f);
    v.y = fmaxf(v.y, 0.0f);
    v.z = fmaxf(v.z, 0.0f);
    v.w = fmaxf(v.w, 0.0f);
    ((float4*)out)[i] = v;
}

extern "C" void kernel_launch(void* const* d_in, const int* in_sizes, int n_in,
                              void* d_out, int out_size, void* d_ws, size_t ws_size,
                              hipStream_t stream) {
    const float* x      = (const float*)d_in[0];   // [N,128]
    const float* W_f    = (const float*)d_in[1];   // [128,128]
    const float* W_b    = (const float*)d_in[2];   // [128,128]
    const float* W_s    = (const float*)d_in[3];   // [128,128]
    const float* drop_u = (const float*)d_in[4];   // [N,128]
    const int*   send   = (const int*)d_in[5];     // [E]
    const int*   recv   = (const int*)d_in[6];     // [E]
    float* out = (float*)d_out;

    const int N = in_sizes[0] / D;   // 100000
    const int E = in_sizes[5];       // 600000

    float* Mf = (float*)d_ws;                       // [N,128]
    float* Mb = (float*)d_ws + (size_t)N * D;       // [N,128]

    const int gemmBlocks = N / 16;   // N = 100000 = 16 * 6250

    // One fused pass: Mf = x@W_f, Mb = x@W_b, out = (x@W_s)*mask; x read once.
    gemm16_wmma_x3<<<gemmBlocks, 128, 0, stream>>>(x, W_f, W_b, W_s, drop_u, Mf, Mb, out);

    // Scatter-accumulate edges: out[r] += Mf[s]; out[s] += Mb[r]
    const long long edgeThreads = (long long)E * 32;
    const int edgeBlocks = (int)((edgeThreads + 255) / 256);
    edge_scatter<<<edgeBlocks, 256, 0, stream>>>(Mf, Mb, send, recv, out, E);

    // ReLU in place
    const long long n4 = (long long)N * D / 4;
    const int reluBlocks = (int)((n4 + 255) / 256);
    relu_inplace<<<reluBlocks, 256, 0, stream>>>(out, n4);
}